// PointNetSetAbstraction_37598143709536
// MI455X (gfx1250) — compile-verified
//
#include <hip/hip_runtime.h>

#define B_  8
#define N_  8192
#define S_  2048      // NPOINT
#define K_  32        // NSAMPLE
#define CF_ 6
#define P_  (B_*S_*K_)   // 524288 columns
#define EPS_ 1e-5f

typedef __attribute__((ext_vector_type(2))) float v2f;
typedef __attribute__((ext_vector_type(8))) float v8f;

// ---------- wave32 argmax (first-max tie-break: lower index) ----------
__device__ __forceinline__ void wave_argmax(float& v, int& i) {
  #pragma unroll
  for (int off = 16; off > 0; off >>= 1) {
    float ov = __shfl_xor(v, off, 32);
    int   oi = __shfl_xor(i, off, 32);
    if (ov > v || (ov == v && oi < i)) { v = ov; i = oi; }
  }
}

// =====================================================================
// 1) Furthest point sampling: one block per batch, coords+dist in regs.
//    Writes new_xyz_t (B,S,3) scratch and new_xyz (B,3,S) into d_out.
// =====================================================================
__global__ void fps_kernel(const float* __restrict__ xyz,
                           float* __restrict__ nxt_xyz,   // (B*S,3)
                           float* __restrict__ out_xyz) { // (B,3,S)
  const int b   = blockIdx.x;
  const int tid = threadIdx.x;         // 0..1023
  const int T   = 1024;
  const int PT  = N_ / T;              // 8 points per thread
  const int lane = tid & 31, wid = tid >> 5;
  __shared__ float redv[32];
  __shared__ int   redi[32];
  __shared__ float bc[3];
  __shared__ int   bidx;

  const float* X = xyz + (size_t)b * 3 * N_;
  float px[PT], py[PT], pz[PT], dl[PT];
  #pragma unroll
  for (int i = 0; i < PT; ++i) {
    int n = tid + i * T;
    px[i] = X[n]; py[i] = X[N_ + n]; pz[i] = X[2 * N_ + n];
    dl[i] = 1e10f;
  }
  if (tid == 0) { bc[0] = X[0]; bc[1] = X[N_]; bc[2] = X[2 * N_]; }
  __syncthreads();

  for (int s = 0; s < S_; ++s) {
    float cx = bc[0], cy = bc[1], cz = bc[2];
    if (tid == 0) {
      float* nt = nxt_xyz + ((size_t)b * S_ + s) * 3;
      nt[0] = cx; nt[1] = cy; nt[2] = cz;
      float* o = out_xyz + (size_t)b * 3 * S_ + s;
      o[0] = cx; o[S_] = cy; o[2 * S_] = cz;
    }
    float bv = -1.f; int bi = 0;
    #pragma unroll
    for (int i = 0; i < PT; ++i) {
      int n = tid + i * T;
      float dx = px[i] - cx, dy = py[i] - cy, dz = pz[i] - cz;
      float d  = dx * dx + dy * dy + dz * dz;
      float dn = fminf(dl[i], d);
      dl[i] = dn;
      if (dn > bv) { bv = dn; bi = n; }   // ascending n -> keeps first max
    }
    wave_argmax(bv, bi);
    if (lane == 0) { redv[wid] = bv; redi[wid] = bi; }
    __syncthreads();
    if (wid == 0) {
      float v = redv[lane]; int i = redi[lane];
      wave_argmax(v, i);
      if (lane == 0) bidx = i;
    }
    __syncthreads();
    int wi = bidx;
    if (tid == (wi & (T - 1))) {
      int slot = wi >> 10;
      float sx = px[0], sy = py[0], sz = pz[0];
      #pragma unroll
      for (int i = 1; i < PT; ++i)
        if (slot == i) { sx = px[i]; sy = py[i]; sz = pz[i]; }
      bc[0] = sx; bc[1] = sy; bc[2] = sz;
    }
    __syncthreads();
  }
}

// =====================================================================
// 2) Ball query + grouping: one wave per centroid; ballot-ordered pick.
//    Writes F (12 x P): rows 0-2 centered xyz, 3-8 features, 9-11 zero.
// =====================================================================
__global__ void ballquery_group_kernel(const float* __restrict__ xyz,
                                       const float* __restrict__ pts,
                                       const float* __restrict__ nxt_xyz,
                                       float* __restrict__ F) {
  const int lane = threadIdx.x & 31;
  const int wl   = threadIdx.x >> 5;
  const int gw   = blockIdx.x * 8 + wl;   // centroid id in [0, B*S)
  const int b    = gw / S_;
  __shared__ int nbr[8][K_];

  const float cx = nxt_xyz[(size_t)gw * 3 + 0];
  const float cy = nxt_xyz[(size_t)gw * 3 + 1];
  const float cz = nxt_xyz[(size_t)gw * 3 + 2];
  const float r2 = (float)(0.4 * 0.4);
  const float* Xb = xyz + (size_t)b * 3 * N_;

  int cnt = 0;
  for (int n0 = 0; n0 < N_ && cnt < K_; n0 += 32) {
    int n = n0 + lane;
    float dx = Xb[n] - cx, dy = Xb[N_ + n] - cy, dz = Xb[2 * N_ + n] - cz;
    bool in = (dx * dx + dy * dy + dz * dz) <= r2;
    unsigned long long m = __ballot(in);
    if (in) {
      int slot = cnt + (int)__popcll(m & ((1ull << lane) - 1ull));
      if (slot < K_) nbr[wl][slot] = n;
    }
    cnt += (int)__popcll(m);
  }
  __syncthreads();
  int j = (lane < cnt) ? nbr[wl][lane] : nbr[wl][0];  // cnt>=1 (center itself)

  size_t col = (size_t)gw * K_ + lane;
  F[0 * (size_t)P_ + col] = Xb[j] - cx;
  F[1 * (size_t)P_ + col] = Xb[N_ + j] - cy;
  F[2 * (size_t)P_ + col] = Xb[2 * N_ + j] - cz;
  const float* Pb = pts + (size_t)b * CF_ * N_;
  #pragma unroll
  for (int c = 0; c < CF_; ++c)
    F[(size_t)(3 + c) * P_ + col] = Pb[(size_t)c * N_ + j];
  #pragma unroll
  for (int r = 9; r < 12; ++r)
    F[(size_t)r * P_ + col] = 0.f;
}

// =====================================================================
// 3) GEMM (Y = W @ X + bias), WMMA f32 16x16x4.
//    One wave owns a full 16-wide column tile and ALL COUT/16 M-tiles:
//    the X panel is loaded exactly once (min HBM traffic), weights are
//    staged transposed in LDS (conflict-free ds loads, zero k-padding).
//    BN=true fuses max(0, x*scale[k]+shift[k]) into the B-operand load.
// =====================================================================
template <int CIN, int KPAD, int COUT, bool BN>
__global__ void gemm_bn_kernel(const float* __restrict__ W,
                               const float* __restrict__ bias,
                               const float* __restrict__ X,
                               const float* __restrict__ scale,
                               const float* __restrict__ shift,
                               float* __restrict__ Y) {
  constexpr int MT = COUT / 16;
  __shared__ float Wt[KPAD][COUT];   // transposed weights, k-padded

  // cooperative staging (once per block; weights are L2-resident)
  for (int i = threadIdx.x; i < KPAD * COUT; i += blockDim.x) {
    int k = i / COUT, m = i % COUT;
    Wt[k][m] = (k < CIN) ? W[m * CIN + k] : 0.f;
  }
  __syncthreads();

  const int lane = threadIdx.x & 31;
  const int nt   = (blockIdx.x * blockDim.x + threadIdx.x) >> 5;  // column tile
  const int half = lane >> 4, l16 = lane & 15;
  const int col  = nt * 16 + l16;

  v8f acc[MT];
  #pragma unroll
  for (int mt = 0; mt < MT; ++mt)
    #pragma unroll
    for (int r = 0; r < 8; ++r)
      acc[mt][r] = bias[mt * 16 + half * 8 + r];

  #pragma unroll
  for (int k0 = 0; k0 < KPAD; k0 += 4) {
    const int ka = k0 + half * 2;
    float b0 = X[(size_t)ka * P_ + col];
    float b1 = X[(size_t)(ka + 1) * P_ + col];
    if (BN) {
      b0 = fmaxf(0.f, fmaf(b0, scale[ka],     shift[ka]));
      b1 = fmaxf(0.f, fmaf(b1, scale[ka + 1], shift[ka + 1]));
    }
    v2f bm; bm.x = b0; bm.y = b1;
    #pragma unroll
    for (int mt = 0; mt < MT; ++mt) {
      v2f a;
      a.x = Wt[ka][mt * 16 + l16];
      a.y = Wt[ka + 1][mt * 16 + l16];
      acc[mt] = __builtin_amdgcn_wmma_f32_16x16x4_f32(false, a, false, bm,
                                                      (short)0, acc[mt],
                                                      false, false);
    }
  }
  #pragma unroll
  for (int mt = 0; mt < MT; ++mt)
    #pragma unroll
    for (int r = 0; r < 8; ++r)
      Y[(size_t)(mt * 16 + half * 8 + r) * P_ + col] = acc[mt][r];
}

// =====================================================================
// 4) Per-channel sum / sumsq for BN batch stats (grid = C * 256 chunks)
// =====================================================================
__global__ void chan_stats_kernel(const float* __restrict__ Y,
                                  float* __restrict__ stats, int C) {
  const int CHUNKS = 256;
  const int c     = blockIdx.x % C;
  const int chunk = blockIdx.x / C;
  const int per   = P_ / CHUNKS;  // 2048
  const float* row = Y + (size_t)c * P_ + (size_t)chunk * per;
  float s = 0.f, s2 = 0.f;
  for (int i = threadIdx.x; i < per; i += blockDim.x) {
    float v = row[i]; s += v; s2 += v * v;
  }
  #pragma unroll
  for (int off = 16; off > 0; off >>= 1) {
    s  += __shfl_xor(s,  off, 32);
    s2 += __shfl_xor(s2, off, 32);
  }
  __shared__ float ls[8], ls2[8];
  const int lane = threadIdx.x & 31, wid = threadIdx.x >> 5;
  if (lane == 0) { ls[wid] = s; ls2[wid] = s2; }
  __syncthreads();
  if (threadIdx.x == 0) {
    float ts = 0.f, t2 = 0.f;
    #pragma unroll
    for (int i = 0; i < 8; ++i) { ts += ls[i]; t2 += ls2[i]; }
    atomicAdd(&stats[2 * c],     ts);
    atomicAdd(&stats[2 * c + 1], t2);
  }
}

__global__ void finalize_stats_kernel(const float* __restrict__ stats,
                                      const float* __restrict__ g,
                                      const float* __restrict__ be,
                                      float* __restrict__ scale,
                                      float* __restrict__ shift, int C) {
  int c = blockIdx.x * blockDim.x + threadIdx.x;
  if (c >= C) return;
  const float invP = 1.f / (float)P_;
  float mu  = stats[2 * c] * invP;
  float var = stats[2 * c + 1] * invP - mu * mu;
  float rs  = rsqrtf(var + EPS_);
  float sc  = rs * g[c];
  scale[c] = sc;
  shift[c] = be[c] - mu * sc;
}

// =====================================================================
// 5) BN3 + ReLU + max over K=32 -> new_points (B,128,S)
// =====================================================================
__global__ void maxpool_kernel(const float* __restrict__ Y3,
                               const float* __restrict__ scale,
                               const float* __restrict__ shift,
                               float* __restrict__ out) {
  const int C = 128;
  int idx = blockIdx.x * blockDim.x + threadIdx.x;  // < B*C*S
  int s = idx % S_;
  int c = (idx / S_) % C;
  int b = idx / (S_ * C);
  const float4* p = (const float4*)(Y3 + (size_t)c * P_ +
                                    ((size_t)(b * S_ + s)) * K_);
  float sc = scale[c], sh = shift[c];
  float m = 0.f;  // ReLU outputs are >= 0
  #pragma unroll
  for (int r = 0; r < K_ / 4; ++r) {
    float4 v = p[r];
    m = fmaxf(m, fmaxf(0.f, fmaf(v.x, sc, sh)));
    m = fmaxf(m, fmaxf(0.f, fmaf(v.y, sc, sh)));
    m = fmaxf(m, fmaxf(0.f, fmaf(v.z, sc, sh)));
    m = fmaxf(m, fmaxf(0.f, fmaf(v.w, sc, sh)));
  }
  out[idx] = m;
}

// =====================================================================
extern "C" void kernel_launch(void* const* d_in, const int* in_sizes, int n_in,
                              void* d_out, int out_size, void* d_ws, size_t ws_size,
                              hipStream_t stream) {
  const float* xyz = (const float*)d_in[0];
  const float* pts = (const float*)d_in[1];
  const float* w1  = (const float*)d_in[2];
  const float* b1  = (const float*)d_in[3];
  const float* g1  = (const float*)d_in[4];
  const float* be1 = (const float*)d_in[5];
  const float* w2  = (const float*)d_in[6];
  const float* b2  = (const float*)d_in[7];
  const float* g2  = (const float*)d_in[8];
  const float* be2 = (const float*)d_in[9];
  const float* w3  = (const float*)d_in[10];
  const float* b3  = (const float*)d_in[11];
  const float* g3  = (const float*)d_in[12];
  const float* be3 = (const float*)d_in[13];

  float* out_xyz = (float*)d_out;                       // (B,3,S)
  float* out_pts = (float*)d_out + (size_t)B_ * 3 * S_; // (B,128,S)

  // ---- workspace layout (Y3 aliases dead F/Y1 region) ----
  char* ws = (char*)d_ws;
  const size_t szY2 = (size_t)64 * P_ * 4;   // 128 MiB
  const size_t szF  = (size_t)12 * P_ * 4;   //  24 MiB
  const size_t szY3 = (size_t)128 * P_ * 4;  // 256 MiB
  float* Y2 = (float*)ws;
  char*  rA = ws + szY2;
  float* F  = (float*)rA;
  float* Y1 = (float*)(rA + szF);
  float* Y3 = (float*)rA;                    // reuses F+Y1 after they die
  char* tail = rA + szY3;
  float* nxt_xyz = (float*)tail;                         // B*S*3 floats
  size_t nxBytes = (((size_t)B_ * S_ * 3 * 4) + 255) & ~(size_t)255;
  float* stats  = (float*)(tail + nxBytes);              // 512 floats
  float* stats1 = stats, *stats2 = stats + 128, *stats3 = stats + 256;
  float* scsh   = stats + 512;                           // 512 floats
  float* sc1 = scsh,       *sh1 = scsh + 64;
  float* sc2 = scsh + 128, *sh2 = scsh + 192;
  float* sc3 = scsh + 256, *sh3 = scsh + 384;

  hipMemsetAsync(stats, 0, 512 * sizeof(float), stream);

  // 1) FPS
  fps_kernel<<<B_, 1024, 0, stream>>>(xyz, nxt_xyz, out_xyz);

  // 2) Ball query + grouping -> F (12 x P)
  ballquery_group_kernel<<<(B_ * S_) / 8, 256, 0, stream>>>(xyz, pts, nxt_xyz, F);

  // GEMM grid: one wave per 16-wide column tile -> P_/16 waves total,
  // 512 threads (16 waves) per block -> exact full waves, EXEC all-ones.
  const int gemm_blocks = (P_ / 16) / 16;   // 2048

  // 3) Layer 1: 9->64 (K padded to 12), no BN on input
  gemm_bn_kernel<9, 12, 64, false><<<gemm_blocks, 512, 0, stream>>>(
      w1, b1, F, nullptr, nullptr, Y1);
  chan_stats_kernel<<<64 * 256, 256, 0, stream>>>(Y1, stats1, 64);
  finalize_stats_kernel<<<1, 64, 0, stream>>>(stats1, g1, be1, sc1, sh1, 64);

  // 4) Layer 2: 64->64, BN1+ReLU fused on load
  gemm_bn_kernel<64, 64, 64, true><<<gemm_blocks, 512, 0, stream>>>(
      w2, b2, Y1, sc1, sh1, Y2);
  chan_stats_kernel<<<64 * 256, 256, 0, stream>>>(Y2, stats2, 64);
  finalize_stats_kernel<<<1, 64, 0, stream>>>(stats2, g2, be2, sc2, sh2, 64);

  // 5) Layer 3: 64->128, BN2+ReLU fused on load (Y3 overwrites dead F/Y1)
  gemm_bn_kernel<64, 64, 128, true><<<gemm_blocks, 512, 0, stream>>>(
      w3, b3, Y2, sc2, sh2, Y3);
  chan_stats_kernel<<<128 * 256, 256, 0, stream>>>(Y3, stats3, 128);
  finalize_stats_kernel<<<1, 128, 0, stream>>>(stats3, g3, be3, sc3, sh3, 128);

  // 6) BN3 + ReLU + max over nsample -> new_points
  maxpool_kernel<<<(B_ * 128 * S_) / 256, 256, 0, stream>>>(Y3, sc3, sh3, out_pts);
}